// PVLoss_29343216566541
// MI455X (gfx1250) — compile-verified
//
#include <hip/hip_runtime.h>
#include <cstdint>

#define BROWS 4096
#define TCOLS 4096
#define THREADS 256
#define NWAVES (THREADS / 32)
#define PER_THREAD (TCOLS / THREADS)   // 16 elements per thread

typedef __attribute__((ext_vector_type(2))) float v2f;
typedef __attribute__((ext_vector_type(8))) float v8f;

__device__ __forceinline__ float wave_reduce_sum(float v) {
#pragma unroll
    for (int o = 16; o > 0; o >>= 1) v += __shfl_down(v, o, 32);
    return v;
}

__device__ __forceinline__ void async_wait_all() {
#if __has_builtin(__builtin_amdgcn_s_wait_asynccnt)
    __builtin_amdgcn_s_wait_asynccnt(0);
#else
    asm volatile("s_wait_asynccnt 0x0" ::: "memory");
#endif
}

// One workgroup per row. Stage preds-row + labels-row in LDS via the CDNA5
// async global->LDS DMA path (single read of HBM), then two passes over LDS.
__global__ __launch_bounds__(THREADS) void pv_row_kernel(
        const float* __restrict__ preds,
        const float* __restrict__ labels,
        float* __restrict__ row_c) {
    __shared__ float smem[2 * TCOLS];        // [0,T): preds row, [T,2T): labels row
    __shared__ float red[8][NWAVES];
    __shared__ float sums[8];
    __shared__ float means[4];

    const int row  = blockIdx.x;
    const int tid  = threadIdx.x;
    const int wid  = tid >> 5;
    const int lane = tid & 31;

    const float* pr = preds  + (size_t)row * TCOLS;
    const float* lr = labels + (size_t)row * TCOLS;

    // ---- async copy: 4 x b128 per array per thread, coalesced across lanes ----
#pragma unroll
    for (int k = 0; k < TCOLS / (THREADS * 4); ++k) {
        const int e = (tid + k * THREADS) * 4;                 // float index in row
        const uint32_t lds_p = (uint32_t)(uintptr_t)(&smem[e]);          // LDS byte addr
        const uint32_t lds_l = (uint32_t)(uintptr_t)(&smem[TCOLS + e]);
        asm volatile("global_load_async_to_lds_b128 %0, %1, off"
                     :: "v"(lds_p), "v"(pr + e) : "memory");
        asm volatile("global_load_async_to_lds_b128 %0, %1, off"
                     :: "v"(lds_l), "v"(lr + e) : "memory");
    }
    async_wait_all();
    __syncthreads();

    // ---- pass 1: peak/valley masked sums & counts (preds and labels) ----
    float v[8];
#pragma unroll
    for (int q = 0; q < 8; ++q) v[q] = 0.0f;

    const int j0 = tid * PER_THREAD;
#pragma unroll
    for (int k = 0; k < PER_THREAD; ++k) {
        const int j = j0 + k;
        if (j >= 1 && j <= TCOLS - 2) {
            const float a = smem[j - 1], x = smem[j], b = smem[j + 1];
            const bool pk = (x > a) && (x > b);
            const bool vl = (x < a) && (x < b);
            v[0] += pk ? x : 0.0f;  v[1] += pk ? 1.0f : 0.0f;
            v[2] += vl ? x : 0.0f;  v[3] += vl ? 1.0f : 0.0f;
            const float a2 = smem[TCOLS + j - 1], x2 = smem[TCOLS + j], b2 = smem[TCOLS + j + 1];
            const bool pk2 = (x2 > a2) && (x2 > b2);
            const bool vl2 = (x2 < a2) && (x2 < b2);
            v[4] += pk2 ? x2 : 0.0f;  v[5] += pk2 ? 1.0f : 0.0f;
            v[6] += vl2 ? x2 : 0.0f;  v[7] += vl2 ? 1.0f : 0.0f;
        }
    }
#pragma unroll
    for (int q = 0; q < 8; ++q) {
        const float r = wave_reduce_sum(v[q]);
        if (lane == 0) red[q][wid] = r;
    }
    __syncthreads();
    if (tid == 0) {
#pragma unroll
        for (int q = 0; q < 8; ++q) {
            float s = 0.0f;
            for (int w = 0; w < NWAVES; ++w) s += red[q][w];
            sums[q] = s;
        }
        means[0] = sums[0] / sums[1];   // preds peak mean
        means[1] = sums[2] / sums[3];   // preds valley mean
        means[2] = sums[4] / sums[5];   // labels peak mean
        means[3] = sums[6] / sums[7];   // labels valley mean
    }
    __syncthreads();

    // ---- pass 2: sbp/dbp conditional masked sums ----
    const float pm  = means[0], vm  = means[1];
    const float pml = means[2], vml = means[3];
#pragma unroll
    for (int q = 0; q < 8; ++q) v[q] = 0.0f;
#pragma unroll
    for (int k = 0; k < PER_THREAD; ++k) {
        const int j = j0 + k;
        if (j >= 1 && j <= TCOLS - 2) {
            const float a = smem[j - 1], x = smem[j], b = smem[j + 1];
            const bool sb = (x > a) && (x > b) && (x >= pm);
            const bool db = (x < a) && (x < b) && (x <= vm);
            v[0] += sb ? x : 0.0f;  v[1] += sb ? 1.0f : 0.0f;
            v[2] += db ? x : 0.0f;  v[3] += db ? 1.0f : 0.0f;
            const float a2 = smem[TCOLS + j - 1], x2 = smem[TCOLS + j], b2 = smem[TCOLS + j + 1];
            const bool sb2 = (x2 > a2) && (x2 > b2) && (x2 >= pml);
            const bool db2 = (x2 < a2) && (x2 < b2) && (x2 <= vml);
            v[4] += sb2 ? x2 : 0.0f;  v[5] += sb2 ? 1.0f : 0.0f;
            v[6] += db2 ? x2 : 0.0f;  v[7] += db2 ? 1.0f : 0.0f;
        }
    }
#pragma unroll
    for (int q = 0; q < 8; ++q) {
        const float r = wave_reduce_sum(v[q]);
        if (lane == 0) red[q][wid] = r;
    }
    __syncthreads();
    if (tid == 0) {
        float s[8];
#pragma unroll
        for (int q = 0; q < 8; ++q) {
            float t = 0.0f;
            for (int w = 0; w < NWAVES; ++w) t += red[q][w];
            s[q] = t;
        }
        const float sbp_p = s[0] / s[1], dbp_p = s[2] / s[3];
        const float sbp_l = s[4] / s[5], dbp_l = s[6] / s[7];
        const float d1 = sbp_p - sbp_l, d2 = dbp_p - dbp_l;
        row_c[row] = d1 * d1 + d2 * d2;
    }
}

// Final reduction of 4096 per-row losses -> scalar mean over (B,2).
// Sum implemented with V_WMMA_F32_16X16X4_F32: with B = all-ones,
// D = A x 1 + C accumulates the 64 A-elements into the tile; total recovered
// from column 0 (lanes 0 and 16). Deterministic, fixed order.
__global__ __launch_bounds__(THREADS) void pv_final_kernel(
        const float* __restrict__ row_c, float* __restrict__ out) {
    __shared__ float wsum[NWAVES];
    const int tid  = threadIdx.x;
    const int wid  = tid >> 5;
    const int lane = tid & 31;

    float wave_total;
#if __has_builtin(__builtin_amdgcn_wmma_f32_16x16x4_f32)
    v8f acc = {0.f, 0.f, 0.f, 0.f, 0.f, 0.f, 0.f, 0.f};
    v2f ones; ones.x = 1.0f; ones.y = 1.0f;
    const int per_wave = BROWS / NWAVES;          // 512 values per wave
#pragma unroll
    for (int it = 0; it < per_wave / 64; ++it) {  // 8 chained WMMAs
        const float* p = row_c + wid * per_wave + it * 64 + lane * 2;
        v2f a; a.x = p[0]; a.y = p[1];
        acc = __builtin_amdgcn_wmma_f32_16x16x4_f32(
            /*neg_a=*/false, a, /*neg_b=*/false, ones,
            /*c_mod=*/(short)0, acc, /*reuse_a=*/false, /*reuse_b=*/false);
    }
    float colsum = 0.0f;
#pragma unroll
    for (int r = 0; r < 8; ++r) colsum += acc[r];
    // column N==0 lives in lane 0 (M=0..7) and lane 16 (M=8..15)
    wave_total = __shfl(colsum, 0, 32) + __shfl(colsum, 16, 32);
#else
    const int per_wave = BROWS / NWAVES;
    float s = 0.0f;
    for (int i = lane; i < per_wave; i += 32) s += row_c[wid * per_wave + i];
    wave_total = wave_reduce_sum(s);
#endif
    if (lane == 0) wsum[wid] = wave_total;
    __syncthreads();
    if (tid == 0) {
        float t = 0.0f;
        for (int w = 0; w < NWAVES; ++w) t += wsum[w];
        out[0] = t * (1.0f / (2.0f * (float)BROWS));
    }
}

extern "C" void kernel_launch(void* const* d_in, const int* in_sizes, int n_in,
                              void* d_out, int out_size, void* d_ws, size_t ws_size,
                              hipStream_t stream) {
    const float* preds  = (const float*)d_in[0];
    const float* labels = (const float*)d_in[1];
    float* row_c = (float*)d_ws;                 // 4096 floats of scratch
    pv_row_kernel<<<BROWS, THREADS, 0, stream>>>(preds, labels, row_c);
    pv_final_kernel<<<1, THREADS, 0, stream>>>(row_c, (float*)d_out);
}